// ContextualRNNCell_56745107915332
// MI455X (gfx1250) — compile-verified
//
#include <hip/hip_runtime.h>

// ContextualRNNCell — CDNA5 (gfx1250) implementation.
// Shapes: B=32, T=64, n=256 (latent), m=128 (input), N=n+m=384.
// All heavy GEMMs use V_WMMA_F32_16X16X4_F32 (fp32 WMMA — required precision
// for the 64-step Riccati-like recurrence with per-step linear solves).
// Per-step working set (~30 MB) is L2-resident (192 MB) -> compute-bound.
// GEMM: each wave produces a 16x64 strip (4 accumulators) with an explicit
// software pipeline (fragments for k+4 loaded before the k-step WMMAs issue).
// Solve: whole 256x256 system staged in LDS (263 KB of the 320 KB/WGP budget,
// stride 257 -> bank-conflict-free), eliminating global-memory round trips.

#define BSZ   32
#define TLEN  64
#define NLAT  256
#define MIN_  128
#define NTOT  384
#define LU_LD 257   // 256+1 pad; 257 % 64 == 1 -> conflict-free column access

typedef __attribute__((ext_vector_type(2))) float v2f;
typedef __attribute__((ext_vector_type(8))) float v8f;

// ---------------------------------------------------------------------------
// Fragment loader for the fp32 WMMA GEMM.
//   TRANSB=0: B is KxN, b = {B[ka][col], B[ka+1][col]} (strided)
//   TRANSB=1: B is NxK, b = {B[col][ka], B[col][ka+1]} (contiguous b64)
// ---------------------------------------------------------------------------
template <int TRANSB>
__device__ __forceinline__ void load_frags(
    const float* __restrict__ Arow, const float* __restrict__ Bb,
    int ka, int col0, int ldb,
    v2f& a, v2f& b0, v2f& b1, v2f& b2, v2f& b3)
{
  a = *(const v2f*)(Arow + ka);
  if (TRANSB) {
    b0 = *(const v2f*)(Bb + (long)(col0     ) * ldb + ka);
    b1 = *(const v2f*)(Bb + (long)(col0 + 16) * ldb + ka);
    b2 = *(const v2f*)(Bb + (long)(col0 + 32) * ldb + ka);
    b3 = *(const v2f*)(Bb + (long)(col0 + 48) * ldb + ka);
  } else {
    const float* Br0 = Bb + (long)ka * ldb;
    const float* Br1 = Br0 + ldb;
    b0.x = Br0[col0];      b0.y = Br1[col0];
    b1.x = Br0[col0 + 16]; b1.y = Br1[col0 + 16];
    b2.x = Br0[col0 + 32]; b2.y = Br1[col0 + 32];
    b3.x = Br0[col0 + 48]; b3.y = Br1[col0 + 48];
  }
}

#define WMMA_F32(acc, a, b) \
  acc = __builtin_amdgcn_wmma_f32_16x16x4_f32(false, a, false, b, (short)0, acc, false, false)

// ---------------------------------------------------------------------------
// Batched fp32 WMMA GEMM:  C[b] = A[b] (MxK) * op(B[b]) (+ bias[col])
// One wave per 16x64 output strip; K stepped by 4 (V_WMMA_F32_16X16X4_F32).
// A-fragment (ISA 7.12.2, 32-bit A 16x4): lanes 0-15 hold K={k0,k0+1},
// lanes 16-31 hold K={k0+2,k0+3}; row = lane&15. B mirrors with col = lane&15.
// C/D: VGPR r -> row = 16*tm + r + 8*(lane>=16), col = 64*tn + 16*j + (lane&15) for acc j.
// Requires: M % 16 == 0, N % 64 == 0, K % 4 == 0, lda/ldb even.
// ---------------------------------------------------------------------------
template <int TRANSB, int HAS_BIAS>
__global__ void wmma_gemm_f32_kernel(
    const float* __restrict__ A, long sA,
    const float* __restrict__ B, long sB,
    float* __restrict__ C, long sC,
    const float* __restrict__ bias,
    int M, int N, int K, int lda, int ldb, int ldc)
{
  const int batch = blockIdx.y;
  const float* Ab = A + (long)batch * sA;
  const float* Bb = B + (long)batch * sB;
  float*       Cb = C + (long)batch * sC;

  const int stripsN = N >> 6;                  // 64-wide strips
  const int strips  = (M >> 4) * stripsN;
  const int wave = blockIdx.x * (blockDim.x >> 5) + (threadIdx.x >> 5);
  if (wave >= strips) return;                  // uniform per wave -> EXEC all-1s

  const int tm = wave / stripsN;
  const int tn = wave % stripsN;
  const int lane = threadIdx.x & 31;
  const int half = lane >> 4;                  // 0: K pair {0,1}, 1: {2,3}
  const int l    = lane & 15;

  const int arow = tm * 16 + l;
  const int col0 = tn * 64 + l;                // this lane's column in strip slot 0
  const float* Arow = Ab + (long)arow * lda;

  v8f acc0 = {}, acc1 = {}, acc2 = {}, acc3 = {};

  // Software pipeline: fragments for step k are in (a,b0..b3) while the
  // fragments for step k+4 are being fetched into (an,c0..c3).
  v2f a, b0, b1, b2, b3;
  load_frags<TRANSB>(Arow, Bb, (half << 1), col0, ldb, a, b0, b1, b2, b3);

#pragma unroll 2
  for (int k0 = 4; k0 < K; k0 += 4) {
    v2f an, c0, c1, c2, c3;
    load_frags<TRANSB>(Arow, Bb, k0 + (half << 1), col0, ldb, an, c0, c1, c2, c3);
    WMMA_F32(acc0, a, b0);
    WMMA_F32(acc1, a, b1);
    WMMA_F32(acc2, a, b2);
    WMMA_F32(acc3, a, b3);
    a = an; b0 = c0; b1 = c1; b2 = c2; b3 = c3;
  }
  WMMA_F32(acc0, a, b0);
  WMMA_F32(acc1, a, b1);
  WMMA_F32(acc2, a, b2);
  WMMA_F32(acc3, a, b3);

  float bv0 = 0.f, bv1 = 0.f, bv2 = 0.f, bv3 = 0.f;
  if (HAS_BIAS) {
    bv0 = bias[col0];
    bv1 = bias[col0 + 16];
    bv2 = bias[col0 + 32];
    bv3 = bias[col0 + 48];
  }

  const int rbase = tm * 16 + (half << 3);
#pragma unroll
  for (int r = 0; r < 8; ++r) {
    float* crow = Cb + (long)(rbase + r) * ldc;
    crow[col0]      = acc0[r] + bv0;
    crow[col0 + 16] = acc1[r] + bv1;
    crow[col0 + 32] = acc2[r] + bv2;
    crow[col0 + 48] = acc3[r] + bv3;
  }
}

// ---------------------------------------------------------------------------
// One-time Gauss-Jordan inversion of W (384x384), storing W_inv^T.
// Single workgroup of 384 threads; W = I + 0.01*noise -> no pivoting needed.
// (384x385 floats = 591 KB exceeds the 320 KB LDS budget, so this one stays
// in global scratch; it runs once, not per step.)
// ---------------------------------------------------------------------------
__global__ void invert_w_kernel(const float* __restrict__ W,
                                float* __restrict__ Winv_t,
                                float* __restrict__ Mw,
                                float* __restrict__ Iw)
{
  const int tid = threadIdx.x;            // 0..383, one row per thread
  __shared__ float rowM[NTOT], rowI[NTOT], fac[NTOT];

  for (int j = 0; j < NTOT; ++j) {
    Mw[(long)tid * NTOT + j] = W[(long)tid * NTOT + j];
    Iw[(long)tid * NTOT + j] = (tid == j) ? 1.0f : 0.0f;
  }
  __syncthreads();

  for (int k = 0; k < NTOT; ++k) {
    const float pinv = 1.0f / Mw[(long)k * NTOT + k];
    const float vm = Mw[(long)k * NTOT + tid] * pinv;
    const float vi = Iw[(long)k * NTOT + tid] * pinv;
    rowM[tid] = vm;
    rowI[tid] = vi;
    fac[tid] = (tid == k) ? 0.0f : Mw[(long)tid * NTOT + k];
    __syncthreads();
    Mw[(long)k * NTOT + tid] = vm;
    Iw[(long)k * NTOT + tid] = vi;
    __syncthreads();
    const float f = fac[tid];
    if (tid != k) {
      for (int j = 0; j < NTOT; ++j) {
        Mw[(long)tid * NTOT + j] -= f * rowM[j];
        Iw[(long)tid * NTOT + j] -= f * rowI[j];
      }
    }
    __syncthreads();
  }
  // Winv_t[i][j] = Winv[j][i]
  for (int j = 0; j < NTOT; ++j)
    Winv_t[(long)tid * NTOT + j] = Iw[(long)j * NTOT + tid];
}

// ---------------------------------------------------------------------------
// State init: A[b] = I (256x256), alpha = 0.
// ---------------------------------------------------------------------------
__global__ void init_state_kernel(float* __restrict__ Amat, float* __restrict__ alpha)
{
  const long total = (long)BSZ * NLAT * NLAT;
  const long stride = (long)gridDim.x * blockDim.x;
  for (long idx = (long)blockIdx.x * blockDim.x + threadIdx.x; idx < total; idx += stride) {
    const int w = (int)(idx & (NLAT * NLAT - 1));
    Amat[idx] = ((w >> 8) == (w & 255)) ? 1.0f : 0.0f;
  }
  for (long idx = (long)blockIdx.x * blockDim.x + threadIdx.x; idx < BSZ * NLAT; idx += stride)
    alpha[idx] = 0.0f;
}

// ---------------------------------------------------------------------------
// Projections: fvec[b] = x_t[b] @ f_w + f_b  (256), beta[b] = x_t[b] @ g_w + g_b (128)
// ---------------------------------------------------------------------------
__global__ void proj_kernel(const float* __restrict__ xt, int xstride,
                            const float* __restrict__ f_w, const float* __restrict__ f_b,
                            const float* __restrict__ g_w, const float* __restrict__ g_b,
                            float* __restrict__ fvec, float* __restrict__ beta)
{
  const int b = blockIdx.y;
  const int o = blockIdx.x * blockDim.x + threadIdx.x;
  if (o >= NTOT) return;
  const float* xv = xt + (long)b * xstride;
  float s = 0.0f;
  if (o < NLAT) {
    for (int k = 0; k < MIN_; ++k) s += xv[k] * f_w[(long)k * NLAT + o];
    fvec[b * NLAT + o] = s + f_b[o];
  } else {
    const int j = o - NLAT;
    for (int k = 0; k < MIN_; ++k) s += xv[k] * g_w[(long)k * MIN_ + j];
    beta[b * MIN_ + j] = s + g_b[j];
  }
}

// ---------------------------------------------------------------------------
// Per-batch Gauss-Jordan solve: alpha[b] += A[b]^{-1} fvec[b].
// One workgroup (256 threads) per batch. The full 256x256 system lives in
// LDS (dynamic, 256*257*4 + 256*4 ~= 264 KB of the 320 KB/WGP budget).
// Row stride 257 (odd) -> column reads across threads hit distinct banks.
// A is only *read* from global (it is still needed for the W11@A GEMM).
// ---------------------------------------------------------------------------
extern __shared__ float solve_smem[];

__global__ void solve_accum_kernel(const float* __restrict__ Amat,
                                   const float* __restrict__ fvec,
                                   float* __restrict__ alpha)
{
  const int b = blockIdx.x;
  const int tid = threadIdx.x;               // 0..255, one row per thread
  const float* Ab = Amat + (long)b * NLAT * NLAT;

  float* lu  = solve_smem;                   // [256][257]
  float* rhs = solve_smem + NLAT * LU_LD;    // [256]

  // Stage A into LDS: thread tid loads column tid of each row (coalesced).
  for (int r = 0; r < NLAT; ++r)
    lu[r * LU_LD + tid] = Ab[(long)r * NLAT + tid];
  rhs[tid] = fvec[b * NLAT + tid];
  __syncthreads();

  for (int k = 0; k < NLAT; ++k) {
    const float pinv = 1.0f / lu[k * LU_LD + k];
    const float v = lu[k * LU_LD + tid] * pinv;       // row k, col tid
    const float f = (tid == k) ? 0.0f : lu[tid * LU_LD + k];  // col k, row tid
    __syncthreads();
    lu[k * LU_LD + tid] = v;                          // write normalized row k
    if (tid == k) rhs[k] *= pinv;
    __syncthreads();
    const float rk = rhs[k];
    if (tid != k) {
      const float* rk_row = lu + k * LU_LD;           // broadcast reads
      float* my_row = lu + tid * LU_LD;
      for (int j = 0; j < NLAT; ++j)
        my_row[j] -= f * rk_row[j];
      rhs[tid] -= f * rk;
    }
    __syncthreads();
  }
  alpha[b * NLAT + tid] += rhs[tid];
}

// ---------------------------------------------------------------------------
// gamma = W_inv_t @ [alpha; beta]; alpha_next = gamma[:256]; y[:,t,:] = gamma[256:].
// ---------------------------------------------------------------------------
__global__ void gamma_kernel(const float* __restrict__ Winv_t,
                             const float* __restrict__ alpha,
                             const float* __restrict__ beta,
                             float* __restrict__ alpha_next,
                             float* __restrict__ y, int t)
{
  const int b = blockIdx.y;
  const int o = blockIdx.x * blockDim.x + threadIdx.x;
  if (o >= NTOT) return;
  const float* wr = Winv_t + (long)o * NTOT;
  const float* av = alpha + b * NLAT;
  const float* bv = beta + b * MIN_;
  float s = 0.0f;
  for (int k = 0; k < NLAT; ++k) s += wr[k] * av[k];
  for (int k = 0; k < MIN_; ++k) s += wr[NLAT + k] * bv[k];
  if (o < NLAT) alpha_next[b * NLAT + o] = s;
  else          y[((long)b * TLEN + t) * MIN_ + (o - NLAT)] = s;
}

// ---------------------------------------------------------------------------
// Host side
// ---------------------------------------------------------------------------
static inline void launch_gemm(hipStream_t s,
                               const float* A, long sA, const float* B, long sB,
                               float* C, long sC, const float* bias,
                               int M, int N, int K, int lda, int ldb, int ldc,
                               int transB, int batches)
{
  const int strips = (M / 16) * (N / 64);
  dim3 grid((strips + 3) / 4, batches);
  if (transB) {
    if (bias) wmma_gemm_f32_kernel<1, 1><<<grid, 128, 0, s>>>(A, sA, B, sB, C, sC, bias, M, N, K, lda, ldb, ldc);
    else      wmma_gemm_f32_kernel<1, 0><<<grid, 128, 0, s>>>(A, sA, B, sB, C, sC, bias, M, N, K, lda, ldb, ldc);
  } else {
    if (bias) wmma_gemm_f32_kernel<0, 1><<<grid, 128, 0, s>>>(A, sA, B, sB, C, sC, bias, M, N, K, lda, ldb, ldc);
    else      wmma_gemm_f32_kernel<0, 0><<<grid, 128, 0, s>>>(A, sA, B, sB, C, sC, bias, M, N, K, lda, ldb, ldc);
  }
}

extern "C" void kernel_launch(void* const* d_in, const int* in_sizes, int n_in,
                              void* d_out, int out_size, void* d_ws, size_t ws_size,
                              hipStream_t stream)
{
  (void)in_sizes; (void)n_in; (void)out_size; (void)ws_size;
  const float* x   = (const float*)d_in[0];   // (B,T,m)
  const float* W   = (const float*)d_in[1];   // (384,384)
  const float* f_w = (const float*)d_in[2];   // (128,256)
  const float* f_b = (const float*)d_in[3];
  const float* g_w = (const float*)d_in[4];   // (128,128)
  const float* g_b = (const float*)d_in[5];
  const float* r_w = (const float*)d_in[6];   // (128,16384)
  const float* r_b = (const float*)d_in[7];
  float* y = (float*)d_out;                   // (B,T,128) f32

  // Workspace carve-up (floats); total ~8.9M floats (~36 MB).
  float* ws = (float*)d_ws;
  float* Winv_t = ws;  ws += (long)NTOT * NTOT;          // 147456
  float* Mw     = ws;  ws += (long)NTOT * NTOT;
  float* Iw     = ws;  ws += (long)NTOT * NTOT;
  float* Abuf0  = ws;  ws += (long)BSZ * NLAT * NLAT;    // 2097152
  float* Abuf1  = ws;  ws += (long)BSZ * NLAT * NLAT;
  float* WU     = ws;  ws += (long)BSZ * NLAT * NTOT;    // 3145728
  float* S      = ws;  ws += (long)BSZ * MIN_ * MIN_;    // 524288
  float* Bmat   = ws;  ws += (long)BSZ * MIN_ * MIN_;
  float* fvec   = ws;  ws += (long)BSZ * NLAT;
  float* betab  = ws;  ws += (long)BSZ * MIN_;
  float* al0    = ws;  ws += (long)BSZ * NLAT;
  float* al1    = ws;  ws += (long)BSZ * NLAT;

  invert_w_kernel<<<1, NTOT, 0, stream>>>(W, Winv_t, Mw, Iw);
  init_state_kernel<<<256, 256, 0, stream>>>(Abuf0, al0);

  float* A  = Abuf0;  float* An = Abuf1;
  float* al = al0;    float* aln = al1;

  const size_t solve_lds = (size_t)(NLAT * LU_LD + NLAT) * sizeof(float); // ~264 KB

  for (int t = 0; t < TLEN; ++t) {
    const float* xt = x + (long)t * MIN_;    // row b at stride T*m = 8192

    // f/g projections
    proj_kernel<<<dim3(3, BSZ), 128, 0, stream>>>(xt, TLEN * MIN_, f_w, f_b,
                                                  g_w, g_b, fvec, betab);
    // alpha += A^{-1} f   (uses old A; LDS-resident Gauss-Jordan)
    solve_accum_kernel<<<BSZ, NLAT, solve_lds, stream>>>(A, fvec, al);

    // S = x_t @ r_w + r_b  : (32x128)@(128x16384), single GEMM over all batches
    launch_gemm(stream, xt, 0, r_w, 0, S, 0, r_b,
                BSZ, MIN_ * MIN_, MIN_, TLEN * MIN_, MIN_ * MIN_, MIN_ * MIN_, 0, 1);

    // Bmat[b] = S[b] @ S[b]^T
    launch_gemm(stream, S, (long)MIN_ * MIN_, S, (long)MIN_ * MIN_,
                Bmat, (long)MIN_ * MIN_, nullptr,
                MIN_, MIN_, MIN_, MIN_, MIN_, MIN_, 1, BSZ);

    // gamma -> y[:,t,:] and alpha_next (reads post-solve alpha)
    gamma_kernel<<<dim3(3, BSZ), 128, 0, stream>>>(Winv_t, al, betab, aln, y, t);

    // WU[:, :256] = W11 @ A[b]   (W11 = W[0:256,0:256], lda=384)
    launch_gemm(stream, W, 0, A, (long)NLAT * NLAT, WU, (long)NLAT * NTOT, nullptr,
                NLAT, NLAT, NLAT, NTOT, NLAT, NTOT, 0, BSZ);
    // WU[:, 256:] = W12 @ Bmat[b] (W12 = W[0:256,256:384])
    launch_gemm(stream, W + NLAT, 0, Bmat, (long)MIN_ * MIN_,
                WU + NLAT, (long)NLAT * NTOT, nullptr,
                NLAT, MIN_, MIN_, NTOT, MIN_, NTOT, 0, BSZ);
    // A_next[b] = WU[b] @ Wt^T   (Wt = W[:256], transB)
    launch_gemm(stream, WU, (long)NLAT * NTOT, W, 0, An, (long)NLAT * NLAT, nullptr,
                NLAT, NLAT, NTOT, NTOT, NTOT, NLAT, 1, BSZ);

    // double-buffer swap (host-side, deterministic)
    float* tp;
    tp = A;  A  = An;  An  = tp;
    tp = al; al = aln; aln = tp;
  }
}